// Custom_Model_Embedding_Group_62277025792621
// MI455X (gfx1250) — compile-verified
//
#include <hip/hip_runtime.h>
#include <hip/hip_bf16.h>

// ---------------------------------------------------------------------------
// Embedding-group pooling:  out[g,d] = sum_n sum_t Wg[t, e[n], d]
// Reordered as            out[g,d] = sum_v count[v] * sum_t Wg[t,v,d]
// => histogram (4MB, u32 atomics) + fully-streaming read of all tables (252MB)
// HBM-bound: ~252MB / 23.3TB/s ~= 11us.  Table reads are non-temporal so the
// count array stays resident in L2.  Wave reductions use v_wmma_f32_16x16x4_f32
// (B==ones "row-sum" trick, layout-robust) -> real WMMA in reduce and gather.
// ---------------------------------------------------------------------------

typedef __attribute__((ext_vector_type(2))) float v2f;
typedef __attribute__((ext_vector_type(8))) float v8f;

// Full 32-lane sum of v using one V_WMMA_F32_16X16X4_F32.
// A slots: {v, 0} per lane (so sum over ALL A entries == wave sum).
// B slots: all 1.0  =>  D[m,n] = sum_k A[m,k]  (row sums, independent of n).
// Sum_m D[m,0] == sum of all A entries == wave sum.  Extraction uses the
// documented C/D layout: VGPR r holds rows m=r (lanes 0-15) and m=r+8
// (lanes 16-31); column n==0 lives at lane 0 and lane 16.
__device__ __forceinline__ float wave_sum_f32(float v) {
    v2f a; a[0] = v;    a[1] = 0.0f;
    v2f b; b[0] = 1.0f; b[1] = 1.0f;
    v8f c = {};
    c = __builtin_amdgcn_wmma_f32_16x16x4_f32(
        /*neg_a=*/false, a, /*neg_b=*/false, b,
        /*c_mod=*/(short)0, c, /*reuse_a=*/false, /*reuse_b=*/false);
    float s = 0.0f;
#pragma unroll
    for (int r = 0; r < 8; ++r) {
        s += __uint_as_float(__builtin_amdgcn_readlane(__float_as_uint(c[r]), 0));
        s += __uint_as_float(__builtin_amdgcn_readlane(__float_as_uint(c[r]), 16));
    }
    return s; // uniform across the wave
}

// ---------------------------------------------------------------------------

__global__ void zero_u32_kernel(unsigned* __restrict__ p, int n) {
    int i = blockIdx.x * blockDim.x + threadIdx.x;
    int stride = gridDim.x * blockDim.x;
    for (; i < n; i += stride) p[i] = 0u;
}

__global__ void hist_kernel(const int* __restrict__ idx,
                            unsigned* __restrict__ cnt, int n) {
    int i = blockIdx.x * blockDim.x + threadIdx.x;
    if (i < n) {
        int v = __builtin_nontemporal_load(idx + i);
        atomicAdd(&cnt[v], 1u);
    }
}

// grid = (NB, 3); one y-slice per table group. Streams all table rows with
// NT loads, FMA with count[v], then wave(WMMA)+block reduction -> partials.
__global__ void reduce_kernel(const float* __restrict__ W0,
                              const float* __restrict__ W1,
                              const float* __restrict__ W2,
                              const unsigned* __restrict__ cnt,
                              float* __restrict__ partials, int V) {
    const int g = blockIdx.y;
    const float* W;
    unsigned rows;
    if (g == 0)      { W = W0; rows = 5u  * (unsigned)V; }
    else if (g == 1) { W = W1; rows = 10u * (unsigned)V; }
    else             { W = W2; rows = 6u  * (unsigned)V; }

    unsigned tid    = blockIdx.x * blockDim.x + threadIdx.x;
    unsigned stride = gridDim.x * blockDim.x;

    float ax = 0.0f, ay = 0.0f, az = 0.0f;
    for (unsigned r = tid; r < rows; r += stride) {
        unsigned v = r % (unsigned)V;                 // table row id
        float c = (float)cnt[v];                      // L2-resident (4MB)
        const float* p = W + (size_t)3 * r;           // streaming, read-once
        float x = __builtin_nontemporal_load(p);
        float y = __builtin_nontemporal_load(p + 1);
        float z = __builtin_nontemporal_load(p + 2);
        ax = fmaf(c, x, ax);
        ay = fmaf(c, y, ay);
        az = fmaf(c, z, az);
    }

    // wave-level reduction via WMMA (EXEC all ones here: full 256-thr blocks,
    // all lanes reconverged after the uniform-bound loop)
    float wx = wave_sum_f32(ax);
    float wy = wave_sum_f32(ay);
    float wz = wave_sum_f32(az);

    __shared__ float sm[8][3];
    int wave = threadIdx.x >> 5;
    int lane = threadIdx.x & 31;
    if (lane == 0) { sm[wave][0] = wx; sm[wave][1] = wy; sm[wave][2] = wz; }
    __syncthreads();
    if (threadIdx.x == 0) {
        float bx = 0.0f, by = 0.0f, bz = 0.0f;
#pragma unroll
        for (int w = 0; w < 8; ++w) { bx += sm[w][0]; by += sm[w][1]; bz += sm[w][2]; }
        float* pp = partials + ((size_t)g * gridDim.x + blockIdx.x) * 3;
        pp[0] = bx; pp[1] = by; pp[2] = bz;
    }
}

// 9 blocks: one per (group, component). Fixed strided-load + tree reduction
// => bitwise deterministic across replays.
__global__ void final_kernel(const float* __restrict__ partials,
                             float* __restrict__ out, int nb) {
    int g = blockIdx.x / 3;
    int comp = blockIdx.x % 3;
    __shared__ float sm[256];
    float s = 0.0f;
    for (int b = threadIdx.x; b < nb; b += 256)
        s += partials[((size_t)g * nb + b) * 3 + comp];
    sm[threadIdx.x] = s;
    __syncthreads();
#pragma unroll
    for (int off = 128; off > 0; off >>= 1) {
        if (threadIdx.x < off) sm[threadIdx.x] += sm[threadIdx.x + off];
        __syncthreads();
    }
    if (threadIdx.x == 0) out[g * 3 + comp] = sm[0];
}

// ---------------------------------------------------------------------------
// Fallback (only if ws_size can't hold the 4MB count array): direct gather
// with WMMA wave reduction + float atomics into d_out.
// ---------------------------------------------------------------------------

__global__ void zero_out_kernel(float* __restrict__ out) {
    if (threadIdx.x < 9) out[threadIdx.x] = 0.0f;
}

__global__ void gather_kernel(const int* __restrict__ e,
                              const float* __restrict__ W0,
                              const float* __restrict__ W1,
                              const float* __restrict__ W2,
                              float* __restrict__ out, int n, int V) {
    int i = blockIdx.x * blockDim.x + threadIdx.x;
    float s[9];
#pragma unroll
    for (int k = 0; k < 9; ++k) s[k] = 0.0f;
    if (i < n) {
        int idx = e[i];
        for (int t = 0; t < 5; ++t) {
            const float* p = W0 + ((size_t)t * V + idx) * 3;
            s[0] += p[0]; s[1] += p[1]; s[2] += p[2];
        }
        for (int t = 0; t < 10; ++t) {
            const float* p = W1 + ((size_t)t * V + idx) * 3;
            s[3] += p[0]; s[4] += p[1]; s[5] += p[2];
        }
        for (int t = 0; t < 6; ++t) {
            const float* p = W2 + ((size_t)t * V + idx) * 3;
            s[6] += p[0]; s[7] += p[1]; s[8] += p[2];
        }
    }
#pragma unroll
    for (int k = 0; k < 9; ++k) {
        float w = wave_sum_f32(s[k]);            // lanes reconverged here
        if ((threadIdx.x & 31) == 0) atomicAdd(&out[k], w);
    }
}

// ---------------------------------------------------------------------------

extern "C" void kernel_launch(void* const* d_in, const int* in_sizes, int n_in,
                              void* d_out, int out_size, void* d_ws, size_t ws_size,
                              hipStream_t stream) {
    const int*   e  = (const int*)d_in[0];
    const float* W0 = (const float*)d_in[1];   // (5,  V, 3)
    const float* W1 = (const float*)d_in[2];   // (10, V, 3)
    const float* W2 = (const float*)d_in[3];   // (6,  V, 3)
    const int N = in_sizes[0];
    const int V = in_sizes[1] / 15;            // 5 * V * 3
    float* out = (float*)d_out;                // 9 floats

    const int NB = 1024;                       // blocks per group in reduce
    size_t cnt_bytes = (size_t)V * sizeof(unsigned);
    size_t part_off  = (cnt_bytes + 255) & ~(size_t)255;
    size_t need      = part_off + (size_t)3 * NB * 3 * sizeof(float);

    if (ws_size >= need && V > 0) {
        unsigned* cnt   = (unsigned*)d_ws;
        float* partials = (float*)((char*)d_ws + part_off);

        zero_u32_kernel<<<(V + 255) / 256, 256, 0, stream>>>(cnt, V);
        hist_kernel<<<(N + 255) / 256, 256, 0, stream>>>(e, cnt, N);
        dim3 grid(NB, 3, 1);
        reduce_kernel<<<grid, 256, 0, stream>>>(W0, W1, W2, cnt, partials, V);
        final_kernel<<<9, 256, 0, stream>>>(partials, out, NB);
    } else {
        zero_out_kernel<<<1, 32, 0, stream>>>(out);
        gather_kernel<<<(N + 255) / 256, 256, 0, stream>>>(e, W0, W1, W2, out, N, V);
    }
}